// LearnedClassVectors_51170240365258
// MI455X (gfx1250) — compile-verified
//
#include <hip/hip_runtime.h>
#include <stdint.h>

typedef float    v2f __attribute__((ext_vector_type(2)));
typedef float    v8f __attribute__((ext_vector_type(8)));
typedef uint32_t v4u __attribute__((ext_vector_type(4)));
typedef int      v8i __attribute__((ext_vector_type(8)));
typedef int      v4i __attribute__((ext_vector_type(4)));

#define N_INT 13
#define VDIM  12

// LDS layout (dynamic shared):
//   [0, 98304)        : 768 x 32 f32 output tile  (row = k channel, col = zw)
//   [98304, 100352)   : 2048 uint8 bucket indices (voxel v = (pz*4+py)*128 + w)
//   [100352, 100976)  : 13 x 12 f32 vectors table (+pad)
//   [100992, 101120)  : 128B dump region for discarded WMMA padding rows
#define TILE_BYTES (768 * 32 * 4)
#define IDX_OFF    TILE_BYTES
#define VEC_OFF    (TILE_BYTES + 2048)
#define DUMP_OFF   (TILE_BYTES + 2048 + 640)
#define SMEM_BYTES (TILE_BYTES + 2048 + 640 + 128)

__launch_bounds__(256)
__global__ void lcv_kernel(const float* __restrict__ x,
                           const float* __restrict__ vec,
                           float* __restrict__ out)
{
    extern __shared__ char smem[];
    float*   tile = (float*)smem;
    uint8_t* sidx = (uint8_t*)(smem + IDX_OFF);
    float*   svec = (float*)(smem + VEC_OFF);
    float*   dump = (float*)(smem + DUMP_OFF);

    const int t   = threadIdx.x;
    const int blk = blockIdx.x;            // b*1024 + zd*32 + zh
    const int zh  = blk & 31;
    const int zd  = (blk >> 5) & 31;
    const int b   = blk >> 10;

    // ---- stage 0: stage embedding table in LDS -------------------------
    if (t < N_INT * VDIM) svec[t] = vec[t];

    // ---- stage A: read x once, bucketize, stash idx bytes in LDS -------
    const float edges[12] = {-1000.f, -900.f, -400.f, -100.f, -50.f, -10.f,
                               20.f,    40.f,   60.f,  100.f, 800.f, 1000.f};
    const float* xb = x + (size_t)b * 2097152 + (size_t)zd * 4 * 16384
                        + (size_t)zh * 4 * 128;
    #pragma unroll
    for (int i = 0; i < 8; ++i) {
        int v   = t + 256 * i;             // (pz*4+py)*128 + w
        int w   = v & 127;
        int row = v >> 7;                  // pz*4+py
        int py  = row & 3;
        int pz  = row >> 2;
        float xv = xb[(size_t)pz * 16384 + py * 128 + w];
        int idx = 0;
        #pragma unroll
        for (int e = 0; e < 12; ++e) idx += (xv >= edges[e]) ? 1 : 0;
        if (idx < 1) idx = 1;              // torch quirk: bucket 0 merged into 1
        sidx[v] = (uint8_t)idx;
    }
    __syncthreads();

    // ---- stage B: WMMA one-hot gather: D = vectors^T x onehot(idx) -----
    // A chunk c (16x4 f32): A[m][k] = vectors[4c+k][m]  (ISA layout: lane=m,
    //   lanes>=16 hold K=2,3; VGPR0 = first K of pair, VGPR1 = second)
    const int lane = t & 31;
    const int m16  = lane & 15;
    const int lohalf = (lane < 16);
    const int kb   = lohalf ? 0 : 2;

    v2f A[4];
    #pragma unroll
    for (int c = 0; c < 4; ++c) {
        int k0 = 4 * c + kb;
        A[c].x = (m16 < VDIM && k0     < N_INT) ? svec[k0 * VDIM + m16]       : 0.f;
        A[c].y = (m16 < VDIM && k0 + 1 < N_INT) ? svec[(k0 + 1) * VDIM + m16] : 0.f;
    }

    const int wave  = t >> 5;
    const int vbase = lohalf ? 0 : 8;
    // per-lane dump slot (byte address) for discarded padding rows v>=12
    float* dumpslot = dump + lane;

    // 128 tasks per block: task = combo*2 + hh ; combo=(pz*4+py)*4+px ; hh = zw half
    #pragma unroll 2
    for (int task = wave; task < 128; task += 8) {
        int hh    = task & 1;
        int combo = task >> 1;
        int px    = combo & 3;
        int row   = combo >> 2;            // pz*4+py
        int n     = m16;                   // output column within half (zw%16)
        int w     = (16 * hh + n) * 4 + px;
        int myidx = (int)sidx[row * 128 + w];

        v8f acc = {0.f, 0.f, 0.f, 0.f, 0.f, 0.f, 0.f, 0.f};
        #pragma unroll
        for (int c = 0; c < 4; ++c) {
            int k0 = 4 * c + kb;
            v2f Bf;
            Bf.x = (myidx == k0)     ? 1.f : 0.f;
            Bf.y = (myidx == k0 + 1) ? 1.f : 0.f;
            acc = __builtin_amdgcn_wmma_f32_16x16x4_f32(
                      false, A[c], false, Bf, (short)0, acc, false, false);
        }

        // D layout: lane l holds column N=l%16; VGPR r holds M=r (lanes<16)
        // or M=r+8 (lanes>=16).  M == v component.  Rows v>=12 are padding:
        // redirect them branchlessly to the dump region (no EXEC juggling).
        float* base = tile + (combo * VDIM + vbase) * 32 + 16 * hh + n;
        #pragma unroll
        for (int r = 0; r < 4; ++r)        // v = vbase+r in [0..11]: always valid
            base[r * 32] = acc[r];
        #pragma unroll
        for (int r = 4; r < 8; ++r) {      // lanes>=16 -> v>=12: dump
            float* p = lohalf ? (base + r * 32) : dumpslot;
            *p = acc[r];
        }
    }
    __syncthreads();

    // ---- stage C: TDM drains the 96KB tile (2D: 768 rows x 128B, strided)
    if (t < 32) {
        uint64_t gaddr = (uint64_t)(uintptr_t)out
                       + 4ull * ((uint64_t)b * 768 * 32768
                               + (uint64_t)zd * 1024 + (uint64_t)zh * 32);
        uint32_t lds_base = (uint32_t)(uintptr_t)tile;   // flat addr low 32 = LDS offset

        // D# group 0: count=1 | lds_addr | global_addr[56:0] | type=2
        v4u g0 = { 1u,
                   lds_base,
                   (uint32_t)gaddr,
                   (uint32_t)((gaddr >> 32) & 0x1FFFFFFu) | (2u << 30) };
        // D# group 1: data_size=4B; tensor_dim0=32; tensor_dim1=768;
        //             tile_dim0=32; tile_dim1=768; tile_dim2=0;
        //             tensor_dim0_stride=32768 elements
        v8i g1 = { (int)0x00020000,        // wg_mask=0, data_size=2 (4B)
                   (int)(32u  << 16),      // [63:48] tensor_dim0 lo16
                   (int)(768u << 16),      // [79:64]=dim0 hi16=0, [95:80] tensor_dim1 lo16
                   (int)(32u  << 16),      // [111:96] dim1 hi16=0, [127:112] tile_dim0
                   (int)768,               // [143:128] tile_dim1, [159:144] tile_dim2=0
                   (int)32768,             // [191:160] tensor_dim0_stride lo32
                   0, 0 };                 // stride0 hi16=0, tensor_dim1_stride=0
        v4i gz4 = {0, 0, 0, 0};            // groups 2/3 unused (2D tile)
        v8i gz8 = {0, 0, 0, 0, 0, 0, 0, 0};
        __builtin_amdgcn_tensor_store_from_lds(g0, g1, gz4, gz4, gz8, 0);
        __builtin_amdgcn_s_wait_tensorcnt(0);
    }
}

extern "C" void kernel_launch(void* const* d_in, const int* in_sizes, int n_in,
                              void* d_out, int out_size, void* d_ws, size_t ws_size,
                              hipStream_t stream) {
    (void)n_in; (void)out_size; (void)d_ws; (void)ws_size;
    const float* x   = (const float*)d_in[0];
    const float* vec = (const float*)d_in[1];
    float* out       = (float*)d_out;

    const int B = in_sizes[0] / (128 * 128 * 128);   // batch (C==1)
    dim3 grid((unsigned)(B * 32 * 32)), block(256);
    hipLaunchKernelGGL(lcv_kernel, grid, block, SMEM_BYTES, stream, x, vec, out);
}